// GraphSage_65498251264494
// MI455X (gfx1250) — compile-verified
//
#include <hip/hip_runtime.h>

typedef __attribute__((ext_vector_type(2))) float v2f;
typedef __attribute__((ext_vector_type(8))) float v8f;

#define FD   128   // feature dim
#define TP   132   // padded LDS row stride (132 % 64 == 4 -> conflict-free column access)
#define EPSN 1e-12f

// ---------------------------------------------------------------- zero scratch
__global__ __launch_bounds__(256) void zero_ws_kernel(float* __restrict__ p, long n) {
  long i = (long)blockIdx.x * blockDim.x + threadIdx.x;
  long stride = (long)gridDim.x * blockDim.x;
  for (; i < n; i += stride) p[i] = 0.0f;
}

// ---------------------------------------------------------------- edge scatter
// One wave32 per edge: 4 floats/lane -> coalesced 512B row gather + 4 f32 atomics/lane.
__global__ __launch_bounds__(256) void edge_scatter_kernel(
    const float* __restrict__ x, const long long* __restrict__ ei,
    const float* __restrict__ ew, float* __restrict__ acc,
    float* __restrict__ cnt, int E) {
  int gw   = (int)((blockIdx.x * blockDim.x + threadIdx.x) >> 5);
  int lane = (int)(threadIdx.x & 31);
  if (gw >= E) return;

  long  src = (long)ei[gw];
  long  dst = (long)ei[(long)E + gw];
  float w   = ew[gw];

  float4 v = *(const float4*)(x + src * FD + lane * 4);
  float* dp = acc + dst * FD + lane * 4;

  __hip_atomic_fetch_add(dp + 0, v.x + w, __ATOMIC_RELAXED, __HIP_MEMORY_SCOPE_AGENT);
  __hip_atomic_fetch_add(dp + 1, v.y + w, __ATOMIC_RELAXED, __HIP_MEMORY_SCOPE_AGENT);
  __hip_atomic_fetch_add(dp + 2, v.z + w, __ATOMIC_RELAXED, __HIP_MEMORY_SCOPE_AGENT);
  __hip_atomic_fetch_add(dp + 3, v.w + w, __ATOMIC_RELAXED, __HIP_MEMORY_SCOPE_AGENT);

  if (lane == 0)
    __hip_atomic_fetch_add(cnt + dst, 1.0f, __ATOMIC_RELAXED, __HIP_MEMORY_SCOPE_AGENT);
}

// ---------------------------------------------------------------- normalize
// One wave32 per 16-node tile. Mean-scale into LDS, then Gram = A*A^T via
// V_WMMA_F32_16X16X4_F32 (A-frag == B-frag for A*A^T under the ISA layouts);
// diagonal of the Gram gives squared row norms. Then ReLU(scale * row).
__global__ __launch_bounds__(32) void node_normalize_kernel(
    const float* __restrict__ acc, const float* __restrict__ cnt,
    float* __restrict__ out, int N) {
  __shared__ float T[16 * TP];
  __shared__ float INV[16];
  __shared__ float SQ[16];

  const int  lane = (int)threadIdx.x;
  const long n0   = (long)blockIdx.x * 16;

  // per-row 1/max(count,1)
  if (lane < 16) {
    long nr = n0 + lane; if (nr > (long)N - 1) nr = (long)N - 1;
    INV[lane] = 1.0f / fmaxf(cnt[nr], 1.0f);
  }
  __syncthreads();

  // load + mean-scale tile into LDS (ds_store_b128, 16B aligned: TP*4 and lane*16 are 16B multiples)
  #pragma unroll
  for (int r = 0; r < 16; ++r) {
    long nr = n0 + r; if (nr > (long)N - 1) nr = (long)N - 1;
    float4 v = *(const float4*)(acc + nr * FD + lane * 4);
    float  s = INV[r];
    float4 sv; sv.x = v.x * s; sv.y = v.y * s; sv.z = v.z * s; sv.w = v.w * s;
    *(float4*)&T[r * TP + lane * 4] = sv;
  }
  __syncthreads();

  // Gram accumulation: D += A(16x4) * A^T(4x16), 32 K-chunks of 4.
  // Layouts: A vgpr0 holds K=kc+2*half (M=lm), vgpr1 K=kc+2*half+1;
  // B vgpr0 holds K=kc+2*half (N=lm) == T[lm][kc+2*half] -> identical fragment.
  const int half = lane >> 4;
  const int lm   = lane & 15;
  v8f d = {0.f, 0.f, 0.f, 0.f, 0.f, 0.f, 0.f, 0.f};
  #pragma unroll
  for (int kc = 0; kc < FD; kc += 4) {
    v2f a = *(const v2f*)&T[lm * TP + kc + 2 * half];
    d = __builtin_amdgcn_wmma_f32_16x16x4_f32(
        /*neg_a=*/false, a, /*neg_b=*/false, a,
        /*c_mod=*/(short)0, d, /*reuse_a=*/false, /*reuse_b=*/false);
  }

  // Extract diagonal D[m][m]:
  //   m in 0..7  -> vgpr m,   lane m        (lanes 0..7)
  //   m in 8..15 -> vgpr m-8, lane m+16     (lanes 24..31)
  {
    int j = lane & 7;
    float val = d[0];
    val = (j == 1) ? d[1] : val;
    val = (j == 2) ? d[2] : val;
    val = (j == 3) ? d[3] : val;
    val = (j == 4) ? d[4] : val;
    val = (j == 5) ? d[5] : val;
    val = (j == 6) ? d[6] : val;
    val = (j == 7) ? d[7] : val;
    if (lane < 8)  SQ[lane]      = val;
    if (lane >= 24) SQ[lane - 16] = val;
  }
  __syncthreads();

  // scale rows by 1/max(norm, eps), ReLU, store (uniform branch keeps EXEC intact)
  #pragma unroll
  for (int r = 0; r < 16; ++r) {
    if (n0 + r < (long)N) {
      float sc = 1.0f / fmaxf(sqrtf(SQ[r]), EPSN);
      float4 v = *(const float4*)&T[r * TP + lane * 4];
      float4 o;
      o.x = fmaxf(v.x * sc, 0.0f);
      o.y = fmaxf(v.y * sc, 0.0f);
      o.z = fmaxf(v.z * sc, 0.0f);
      o.w = fmaxf(v.w * sc, 0.0f);
      *(float4*)(out + (n0 + r) * FD + lane * 4) = o;
    }
  }
}

// ---------------------------------------------------------------- launch
extern "C" void kernel_launch(void* const* d_in, const int* in_sizes, int n_in,
                              void* d_out, int out_size, void* d_ws, size_t ws_size,
                              hipStream_t stream) {
  const float*     x  = (const float*)d_in[0];
  const long long* ei = (const long long*)d_in[1];   // int64 edge_index (2,E)
  const float*     ew = (const float*)d_in[2];       // (E,1)

  const int N = in_sizes[0] / FD;   // 50000
  const int E = in_sizes[2];        // 625000

  float* acc = (float*)d_ws;                 // N*128 f32 accumulator (L2-resident)
  float* cnt = acc + (size_t)N * FD;         // N f32 counts (contiguous after acc)

  // 1) zero accumulator + counts (harness poisons ws)
  {
    long n = (long)N * FD + N;
    int blocks = 1024;
    zero_ws_kernel<<<blocks, 256, 0, stream>>>(acc, n);
  }

  // 2) scatter-add: one wave per edge, 8 waves per 256-thread block
  {
    int blocks = (E + 7) / 8;
    edge_scatter_kernel<<<blocks, 256, 0, stream>>>(x, ei, ew, acc, cnt, E);
  }

  // 3) mean + WMMA Gram row-norm + ReLU: one wave per 16-node tile
  {
    int blocks = (N + 15) / 16;
    node_normalize_kernel<<<blocks, 32, 0, stream>>>(acc, cnt, (float*)d_out, N);
  }
}